// WarpAlignmentBlock_63393717289431
// MI455X (gfx1250) — compile-verified
//
#include <hip/hip_runtime.h>
#include <hip/hip_bf16.h>
#include <cmath>

typedef __attribute__((ext_vector_type(2))) float v2f;
typedef __attribute__((ext_vector_type(8))) float v8f;

constexpr int Bn = 8;
constexpr int Cn = 256;
constexpr int Hn = 128;
constexpr int Wn = 128;
constexpr int HW = Hn * Wn;          // 16384

// ---------------------------------------------------------------------------
// Kernel 1: flow = tanh(x[b,:,h,w] . conv_w^T + conv_b) via fp32 WMMA.
// Each wave handles 32 pixels (two 16-pixel M-tiles, shared B fragment).
// A tile: 16(M=pixels) x 4(K=channels) fp32 -> v2f per lane
//   lanes 0-15 : M=lane,    VGPR0=K(kb+0), VGPR1=K(kb+1)
//   lanes 16-31: M=lane-16, VGPR0=K(kb+2), VGPR1=K(kb+3)
// B tile: 4(K) x 16(N) fp32, N=0..1 are conv_w rows, N>=2 zero-padded in LDS.
// D: 16x16 fp32 accumulator (v8f); columns 0,1 are the two flow components.
// ---------------------------------------------------------------------------
__global__ __launch_bounds__(256) void flow_wmma_kernel(
    const float* __restrict__ x,       // (B, C, H, W)
    const float* __restrict__ conv_w,  // (2, C)
    const float* __restrict__ conv_b,  // (2,)
    float* __restrict__ flow)          // (B, 2, H, W)  [workspace]
{
    // 16 x 256 B-operand staging: rows 0,1 = conv_w, rows 2..15 = 0.
    __shared__ float sw[16 * 256];
    for (int i = threadIdx.x; i < 16 * 256; i += 256)
        sw[i] = (i < 2 * 256) ? conv_w[i] : 0.0f;
    __syncthreads();

    const int lane  = threadIdx.x & 31;
    const int wv    = threadIdx.x >> 5;
    const int tile  = blockIdx.x * 8 + wv;   // 32 pixels per wave
    const int Pbase = tile * 32;
    const int b     = Pbase / HW;
    const int p0    = Pbase % HW;            // HW % 32 == 0, never crosses batch

    const int mrow = lane & 15;              // M (pixels) / N (outputs) index
    const int koff = (lane >> 4) << 1;       // 0 for lanes 0-15, 2 for 16-31

    const float* xb  = x + (size_t)b * Cn * HW + (size_t)(p0 + mrow);
    const float* swb = &sw[mrow * 256 + koff];

    v8f acc0 = {};
    v8f acc1 = {};

#pragma unroll 4
    for (int kb = 0; kb < Cn; kb += 4) {
        const float* xp = xb + (size_t)(kb + koff) * HW;
        v2f a0, a1, bb;
        a0.x = __builtin_nontemporal_load(xp);            // pixels p0+m,  K=kb+koff
        a0.y = __builtin_nontemporal_load(xp + HW);       //               K=kb+koff+1
        a1.x = __builtin_nontemporal_load(xp + 16);       // pixels p0+16+m
        a1.y = __builtin_nontemporal_load(xp + HW + 16);
        bb.x = swb[kb];                                   // conv_w[n][kb+koff]
        bb.y = swb[kb + 1];
        acc0 = __builtin_amdgcn_wmma_f32_16x16x4_f32(
                   false, a0, false, bb, (short)0, acc0, false, false);
        acc1 = __builtin_amdgcn_wmma_f32_16x16x4_f32(
                   false, a1, false, bb, (short)0, acc1, false, false);
    }

    // D layout: lanes 0-15 -> N=lane, VGPR r -> M=r; lanes 16-31 -> M=r+8.
    const int n  = lane & 15;
    const int mb = (lane >> 4) * 8;
    if (n < 2) {
        const float bias = conv_b[n];
        float* f0 = flow + (size_t)(b * 2 + n) * HW + p0;
#pragma unroll
        for (int r = 0; r < 8; r++) {
            f0[mb + r]      = tanhf(acc0[r] + bias);
            f0[mb + r + 16] = tanhf(acc1[r] + bias);
        }
    }
}

// ---------------------------------------------------------------------------
// Kernel 2: bilinear warp with zero padding, align_corners=True.
// px = w + dx*(W-1)/2 ; py = h + dy*(H-1)/2.
// Block = 256 pixels; grid.y selects a 32-channel chunk so index/weight math
// is amortized over 32 gathers per thread.
// ---------------------------------------------------------------------------
__global__ __launch_bounds__(256) void warp_kernel(
    const float* __restrict__ fm,    // (B, C, H, W)
    const float* __restrict__ flow,  // (B, 2, H, W)
    float* __restrict__ out)         // (B, C, H, W)
{
    const int P  = blockIdx.x * 256 + threadIdx.x;  // global pixel
    const int b  = P / HW;
    const int p  = P % HW;
    const int h  = p / Wn;
    const int w  = p % Wn;
    const int c0 = blockIdx.y * 32;

    const float dx = flow[(size_t)(b * 2 + 0) * HW + p];
    const float dy = flow[(size_t)(b * 2 + 1) * HW + p];

    const float px = (float)w + dx * (0.5f * (float)(Wn - 1));
    const float py = (float)h + dy * (0.5f * (float)(Hn - 1));

    const float x0f = floorf(px), y0f = floorf(py);
    const float wx = px - x0f,   wy = py - y0f;
    const int x0 = (int)x0f, y0 = (int)y0f;
    const int x1 = x0 + 1,   y1 = y0 + 1;

    const bool vx0 = (x0 >= 0) & (x0 < Wn);
    const bool vx1 = (x1 >= 0) & (x1 < Wn);
    const bool vy0 = (y0 >= 0) & (y0 < Hn);
    const bool vy1 = (y1 >= 0) & (y1 < Hn);

    const int xc0 = min(max(x0, 0), Wn - 1);
    const int xc1 = min(max(x1, 0), Wn - 1);
    const int yc0 = min(max(y0, 0), Hn - 1);
    const int yc1 = min(max(y1, 0), Hn - 1);

    const float w00 = (1.0f - wx) * (1.0f - wy) * (float)(vx0 & vy0);
    const float w01 = wx * (1.0f - wy)          * (float)(vx1 & vy0);
    const float w10 = (1.0f - wx) * wy          * (float)(vx0 & vy1);
    const float w11 = wx * wy                   * (float)(vx1 & vy1);

    const int o00 = yc0 * Wn + xc0;
    const int o01 = yc0 * Wn + xc1;
    const int o10 = yc1 * Wn + xc0;
    const int o11 = yc1 * Wn + xc1;

    const float* fp = fm  + (size_t)(b * Cn + c0) * HW;
    float*       op = out + (size_t)(b * Cn + c0) * HW + p;

#pragma unroll 4
    for (int ci = 0; ci < 32; ci++) {
        float v = w00 * fp[o00] + w01 * fp[o01]
                + w10 * fp[o10] + w11 * fp[o11];
        __builtin_nontemporal_store(v, op);
        fp += HW;
        op += HW;
    }
}

extern "C" void kernel_launch(void* const* d_in, const int* in_sizes, int n_in,
                              void* d_out, int out_size, void* d_ws, size_t ws_size,
                              hipStream_t stream) {
    const float* x  = (const float*)d_in[0];  // (8,256,128,128)
    const float* fm = (const float*)d_in[1];  // (8,256,128,128)
    const float* cw = (const float*)d_in[2];  // (2,256)
    const float* cb = (const float*)d_in[3];  // (2,)
    float* out  = (float*)d_out;
    float* flow = (float*)d_ws;               // 8*2*128*128 floats = 1 MB

    // Kernel 1: 131072 pixels / (8 waves * 32 pixels) = 512 blocks.
    flow_wmma_kernel<<<dim3(Bn * HW / 256), 256, 0, stream>>>(x, cw, cb, flow);

    // Kernel 2: 512 pixel-blocks x 8 channel-chunks.
    warp_kernel<<<dim3(Bn * HW / 256, Cn / 32), 256, 0, stream>>>(fm, flow, out);
}